// With_other_joints_9405978378880
// MI455X (gfx1250) — compile-verified
//
#include <hip/hip_runtime.h>
#include <math.h>

// Problem constants (match reference)
#define G_    5
#define B_    256
#define CIN   24
#define T_    2048
#define KOUT  3
#define KW    3
#define LOUT  2046          // T - KW + 1
#define KDIM  72            // CIN * KW
#define NKST  18            // KDIM / 4  (WMMA f32 K-step = 4)
#define EPSBN 1e-5f

typedef __attribute__((ext_vector_type(2))) float v2f;
typedef __attribute__((ext_vector_type(8))) float v8f;

__global__ void zero_stats_kernel(float* __restrict__ stats) {
    if (threadIdx.x < 32) stats[threadIdx.x] = 0.0f;
}

// Pass 1: conv via V_WMMA_F32_16X16X4_F32 + bias, store y to ws, accumulate stats.
// One block per (g,b); 8 waves; each wave owns 16 l-tiles of width 16 (128 tiles total).
__global__ __launch_bounds__(256) void conv_wmma_stats_kernel(
    const float* __restrict__ x,      // (G,B,CIN,T)
    const float* __restrict__ w,      // (G,KOUT,CIN,KW) -> [g*216 + k*72 + cj]
    const float* __restrict__ bias,   // (G,KOUT)
    float* __restrict__ yws,          // (G*B, LOUT, 3)
    float* __restrict__ stats)        // (G,KOUT,2) sum/sumsq
{
    const int gb   = blockIdx.x;
    const int g    = gb / B_;
    const int lane = threadIdx.x & 31;
    const int wave = threadIdx.x >> 5;
    const int half = lane >> 4;       // 0: lanes 0-15, 1: lanes 16-31
    const int m    = lane & 15;       // row (A/D) or column (B) index

    const float* wg = w + g * (KOUT * KDIM);
    const float* xb = x + (size_t)gb * (CIN * T_);
    float*       yb = yws + (size_t)gb * (LOUT * 3);

    // A fragments (weights). A[16x4]: lane m holds row m, K = 4*kk + 2*half + v.
    // Rows m>=3 of D are never consumed, so clamp the row instead of zero-padding:
    // unconditional loads, no exec-mask branches.
    const int mrow = (m < KOUT) ? m : (KOUT - 1);
    const float* wrow = wg + mrow * KDIM + 2 * half;
    v2f afrag[NKST];
#pragma unroll
    for (int kk = 0; kk < NKST; ++kk) {
        v2f af;
        af.x = wrow[4 * kk + 0];
        af.y = wrow[4 * kk + 1];
        afrag[kk] = af;
    }

    const float b0 = bias[g * KOUT + 0];
    const float b1 = bias[g * KOUT + 1];
    const float b2 = bias[g * KOUT + 2];

    float ls0 = 0.f, ls1 = 0.f, ls2 = 0.f;   // local sums
    float lq0 = 0.f, lq1 = 0.f, lq2 = 0.f;   // local sum-of-squares

    for (int i = 0; i < 16; ++i) {
        const int l0 = (wave + 8 * i) * 16;
        const int l  = l0 + m;                        // this lane's column for B / D
        const bool valid = (l < LOUT);                // tail tile: 2 dead columns
        const int lc = valid ? l : (LOUT - 1);        // clamp: keeps l+j <= 2047 in-bounds
        const float* xl = xb + lc;                    // per-lane base; constant offsets below

        v8f acc = {};
#pragma unroll
        for (int kk = 0; kk < NKST; ++kk) {
            // B[4x16]: lane n=m holds column, K = 4*kk + 2*half + v.
            // X[cj][l] = x[cj/3][l + cj%3]; all offsets fold to immediates.
            const int kA      = 4 * kk;
            const int offH0v0 = ((kA + 0) / 3) * T_ + ((kA + 0) % 3);
            const int offH0v1 = ((kA + 1) / 3) * T_ + ((kA + 1) % 3);
            const int offH1v0 = ((kA + 2) / 3) * T_ + ((kA + 2) % 3);
            const int offH1v1 = ((kA + 3) / 3) * T_ + ((kA + 3) % 3);
            const int o0 = half ? offH1v0 : offH0v0;
            const int o1 = half ? offH1v1 : offH0v1;
            v2f bf;
            bf.x = xl[o0];
            bf.y = xl[o1];
            acc = __builtin_amdgcn_wmma_f32_16x16x4_f32(
                      false, afrag[kk], false, bf, (short)0, acc, false, false);
        }

        // D layout: VGPR v, lanes 0-15 hold row M=v, N=lane. Rows 0..2 are y.
        if (half == 0 && valid) {
            const float y0 = acc[0] + b0;
            const float y1 = acc[1] + b1;
            const float y2 = acc[2] + b2;
            const size_t off = (size_t)l * 3;
            yb[off + 0] = y0; yb[off + 1] = y1; yb[off + 2] = y2;
            ls0 += y0; ls1 += y1; ls2 += y2;
            lq0 += y0 * y0; lq1 += y1 * y1; lq2 += y2 * y2;
        }
    }

    // Block reduction via LDS float atomics, then 6 global atomics per block.
    __shared__ float ssum[3], ssq[3];
    if (threadIdx.x < 3) { ssum[threadIdx.x] = 0.f; ssq[threadIdx.x] = 0.f; }
    __syncthreads();
    if (half == 0) {
        atomicAdd(&ssum[0], ls0); atomicAdd(&ssum[1], ls1); atomicAdd(&ssum[2], ls2);
        atomicAdd(&ssq[0],  lq0); atomicAdd(&ssq[1],  lq1); atomicAdd(&ssq[2],  lq2);
    }
    __syncthreads();
    if (threadIdx.x < 3) {
        atomicAdd(&stats[(g * KOUT + threadIdx.x) * 2 + 0], ssum[threadIdx.x]);
        atomicAdd(&stats[(g * KOUT + threadIdx.x) * 2 + 1], ssq[threadIdx.x]);
    }
}

// Pass 2: batchnorm + sigmoid + 8-way channel broadcast into (B, G*8, LOUT, 3).
__global__ __launch_bounds__(256) void bn_sig_expand_kernel(
    const float* __restrict__ yws,    // (G*B, LOUT, 3)
    const float* __restrict__ stats,  // (G,KOUT,2)
    const float* __restrict__ gamma,  // (G,KOUT)
    const float* __restrict__ beta,   // (G,KOUT)
    float* __restrict__ out)          // (B, G*8, LOUT, 3)
{
    const int idx = blockIdx.x * 256 + threadIdx.x;   // over G*B*LOUT (exact multiple of 256)
    const int l  = idx % LOUT;
    const int gb = idx / LOUT;
    const int b  = gb % B_;
    const int g  = gb / B_;

    const float invN = 1.0f / (float)(B_ * LOUT);
    float sc[3], sh[3];
#pragma unroll
    for (int k = 0; k < 3; ++k) {
        const float s1  = stats[(g * KOUT + k) * 2 + 0];
        const float s2  = stats[(g * KOUT + k) * 2 + 1];
        const float mu  = s1 * invN;
        const float var = fmaxf(s2 * invN - mu * mu, 0.0f);
        const float inv = rsqrtf(var + EPSBN);
        const float gm  = gamma[g * KOUT + k];
        const float bt  = beta[g * KOUT + k];
        sc[k] = gm * inv;
        sh[k] = bt - mu * gm * inv;
    }

    const size_t yoff = (size_t)idx * 3;
    const float y0 = yws[yoff + 0];
    const float y1 = yws[yoff + 1];
    const float y2 = yws[yoff + 2];
    const float s0 = 1.0f / (1.0f + expf(-(y0 * sc[0] + sh[0])));
    const float s1 = 1.0f / (1.0f + expf(-(y1 * sc[1] + sh[1])));
    const float s2 = 1.0f / (1.0f + expf(-(y2 * sc[2] + sh[2])));

    const size_t base = ((size_t)(b * (G_ * 8) + g * 8) * LOUT + l) * 3;
#pragma unroll
    for (int c = 0; c < 8; ++c) {
        const size_t o = base + (size_t)c * (LOUT * 3);
        out[o + 0] = s0; out[o + 1] = s1; out[o + 2] = s2;
    }
}

extern "C" void kernel_launch(void* const* d_in, const int* in_sizes, int n_in,
                              void* d_out, int out_size, void* d_ws, size_t ws_size,
                              hipStream_t stream) {
    const float* x_other = (const float*)d_in[0];
    // d_in[1] = x_self: only its shape (cs=8) matters; values unused.
    const float* w     = (const float*)d_in[2];
    const float* bias  = (const float*)d_in[3];
    const float* gamma = (const float*)d_in[4];
    const float* beta  = (const float*)d_in[5];
    float* out = (float*)d_out;

    float* stats = (float*)d_ws;            // 32 floats (zeroed each call)
    float* yws   = (float*)d_ws + 32;       // G*B*LOUT*3 floats = 31.4 MB

    zero_stats_kernel<<<1, 32, 0, stream>>>(stats);
    conv_wmma_stats_kernel<<<G_ * B_, 256, 0, stream>>>(x_other, w, bias, yws, stats);
    const int total = G_ * B_ * LOUT;       // 2,618,880 = 10230 * 256
    bn_sig_expand_kernel<<<total / 256, 256, 0, stream>>>(yws, stats, gamma, beta, out);
}